// HSMM_12558484374294
// MI455X (gfx1250) — compile-verified
//
#include <hip/hip_runtime.h>
#include <hip/hip_bf16.h>

// ---------------------------------------------------------------------------
// HSMM forward (CDNA5 / gfx1250, wave32)
//   S=8 states, D=10 durations, C=64 features, B=4, T=2048, H=W=8
//
// Kernel 1: per-pixel 1x1 convs + softmaxes (trans, dur, mu, m2)   [tiny]
// Kernel 2: emission om = obs(T x C) * mu^T(C x S) via V_WMMA_F32_16X16X4_F32
//           with DOUBLE-BUFFERED GLOBAL_LOAD_ASYNC_TO_LDS_B128 staging
//           (ASYNCcnt-tracked DMA overlapped with WMMA compute)
// Kernel 3: sequential alpha scan (8 lanes per pixel, 4 pixels per wave),
//           fused state_probs normalization + final logsumexp, with
//           global_prefetch_b8 ahead of the serial chain
// ---------------------------------------------------------------------------

#define S_ 8
#define D_ 10
#define C_ 64
#define B_ 4
#define T_ 2048
#define HW_ 64
#define P_ (B_ * HW_)   // 256 pixels

typedef __attribute__((ext_vector_type(2))) float v2f;
typedef __attribute__((ext_vector_type(8))) float v8f;

// ---------------------------------------------------------------------------
// Kernel 1: per-pixel precompute.  One block per pixel, 64 threads.
// ---------------------------------------------------------------------------
__global__ void hsmm_precompute(const float* __restrict__ features,
                                const float* __restrict__ trans_w,
                                const float* __restrict__ trans_b,
                                const float* __restrict__ dur_w,
                                const float* __restrict__ dur_b,
                                const float* __restrict__ emis_w,
                                const float* __restrict__ emis_b,
                                float* __restrict__ dur_out,   // (B,S,D,H,W)
                                float* __restrict__ trans_ws,  // [p][from][to]
                                float* __restrict__ mu_ws,     // [p][s][c]
                                float* __restrict__ m2_ws)     // [p][s]
{
    const int p   = blockIdx.x;        // pixel
    const int b   = p >> 6;
    const int hw  = p & 63;
    const int tid = threadIdx.x;       // 0..63

    __shared__ float sf[C_];
    __shared__ float st[S_ * S_];
    __shared__ float sd[S_ * D_];
    __shared__ float smu[S_ * C_];

    sf[tid] = features[(b * C_ + tid) * HW_ + hw];
    __syncthreads();

    // ---- transition logits: 64 outputs, one per thread ----
    {
        float acc = trans_b[tid];
        #pragma unroll 8
        for (int c = 0; c < C_; ++c) acc += trans_w[tid * C_ + c] * sf[c];
        st[tid] = acc;
    }

    // ---- duration logits: 80 outputs ----
    for (int o = tid; o < S_ * D_; o += 64) {
        float acc = dur_b[o];
        #pragma unroll 8
        for (int c = 0; c < C_; ++c) acc += dur_w[o * C_ + c] * sf[c];
        sd[o] = acc;
    }

    // ---- emission means mu: 512 outputs, 8 per thread ----
    #pragma unroll
    for (int i = 0; i < S_; ++i) {
        const int o = i * C_ + tid;    // s = i, c = tid
        float acc = emis_b[o];
        #pragma unroll 8
        for (int c = 0; c < C_; ++c) acc += emis_w[o * C_ + c] * sf[c];
        smu[o] = acc;
        mu_ws[(p * S_ + i) * C_ + tid] = acc;
    }
    __syncthreads();

    if (tid < S_) {
        const int s = tid;
        // softmax over destination state (trans row s)
        {
            float m = -3.0e38f;
            for (int j = 0; j < S_; ++j) m = fmaxf(m, st[s * S_ + j]);
            float sum = 0.f, e[S_];
            for (int j = 0; j < S_; ++j) { e[j] = __expf(st[s * S_ + j] - m); sum += e[j]; }
            const float inv = 1.0f / sum;
            for (int j = 0; j < S_; ++j)
                trans_ws[p * (S_ * S_) + s * S_ + j] = e[j] * inv;
        }
        // softmax over duration (dur row s)
        {
            float m = -3.0e38f;
            for (int d = 0; d < D_; ++d) m = fmaxf(m, sd[s * D_ + d]);
            float sum = 0.f, e[D_];
            for (int d = 0; d < D_; ++d) { e[d] = __expf(sd[s * D_ + d] - m); sum += e[d]; }
            const float inv = 1.0f / sum;
            for (int d = 0; d < D_; ++d)
                dur_out[((b * S_ + s) * D_ + d) * HW_ + hw] = e[d] * inv;
        }
        // m2[s] = sum_c mu^2
        {
            float acc = 0.f;
            for (int c = 0; c < C_; ++c) { float v = smu[s * C_ + c]; acc += v * v; }
            m2_ws[p * S_ + s] = acc;
        }
    }
}

// ---------------------------------------------------------------------------
// Kernel 2: emission scores via WMMA + double-buffered async LDS staging.
//   grid = (T/16, B), block = 256 threads (8 waves).
//   Chunk = 4 channels x 16 t x 64 pixels, staged via
//   GLOBAL_LOAD_ASYNC_TO_LDS_B128 (ASYNCcnt); next chunk's DMA overlaps
//   this chunk's V_WMMA_F32_16X16X4_F32 work.
// ---------------------------------------------------------------------------
#define CHUNK_C 4
#define NCHUNK (C_ / CHUNK_C)   // 16
#define LDS_PITCH 68            // 64 pixels + pad; keeps 16B alignment

__global__ void hsmm_emission(const float* __restrict__ observations,
                              const float* __restrict__ mu_ws,
                              const float* __restrict__ m2_ws,
                              float* __restrict__ emis_ws)  // [p][t][s]
{
    const int t0 = blockIdx.x * 16;
    const int b  = blockIdx.y;
    const int tid  = threadIdx.x;
    const int wave = tid >> 5;
    const int lane = tid & 31;
    const int tl   = lane & 15;          // A-frag row (t within tile)
    const int n    = lane & 15;          // B/D column (state, padded to 16)
    const bool lowHalf = (lane < 16);

    __shared__ __align__(16) float lds[2][CHUNK_C * 16 * LDS_PITCH];

    const unsigned long long obs_base = (unsigned long long)(uintptr_t)observations;
    // loader mapping: thread covers (q = float4 along hw, tt = t row); it = channel
    const int q  = tid & 15;
    const int tt = (tid >> 4) & 15;

    // Issue one chunk's async DMA: 4 wave-instructions, ASYNCcnt += 4.
    auto issue = [&](int chunk, int buf) {
        #pragma unroll
        for (int it = 0; it < CHUNK_C; ++it) {
            const unsigned int goff = (unsigned int)(
                (((b * C_ + chunk * CHUNK_C + it) * T_ + t0 + tt) * 64 + q * 4) * 4);
            const unsigned int ldsb = (unsigned int)(uintptr_t)
                &lds[buf][(it * 16 + tt) * LDS_PITCH + q * 4];
            asm volatile("global_load_async_to_lds_b128 %0, %1, %2"
                         :: "v"(ldsb), "v"(goff), "s"(obs_base)
                         : "memory");
        }
    };

    v8f   acc[8];
    float o2p[8];
    #pragma unroll
    for (int j = 0; j < 8; ++j) {
        acc[j] = (v8f)0.0f;
        o2p[j] = 0.0f;
    }

    issue(0, 0);
    for (int chunk = 0; chunk < NCHUNK; ++chunk) {
        const int cur = chunk & 1;
        if (chunk + 1 < NCHUNK) {
            issue(chunk + 1, cur ^ 1);                    // prefetch next chunk
            asm volatile("s_wait_asynccnt 4" ::: "memory"); // drain current only
        } else {
            asm volatile("s_wait_asynccnt 0" ::: "memory");
        }
        __syncthreads();   // all waves' DMA for `cur` visible

        const float* L = lds[cur];
        #pragma unroll
        for (int j = 0; j < 8; ++j) {
            const int px = wave * 8 + j;        // pixel hw
            const int p  = b * HW_ + px;        // global pixel
            // ---- A fragment (16x4 f32): lanes 0-15 hold K=0,1; 16-31 K=2,3
            const int kh = lowHalf ? 0 : 2;
            v2f a;
            a.x = L[((kh + 0) * 16 + tl) * LDS_PITCH + px];
            a.y = L[((kh + 1) * 16 + tl) * LDS_PITCH + px];
            o2p[j] += a.x * a.x + a.y * a.y;    // accumulate ||obs_t||^2

            // ---- B fragment (4x16 f32): B[k][n] = mu[n][k], zero-pad n>=8
            v2f bf; bf.x = 0.0f; bf.y = 0.0f;
            if (n < S_) {
                const int basei = (p * S_ + n) * C_ + chunk * CHUNK_C;
                bf.x = mu_ws[basei + (lowHalf ? 0 : 1)];
                bf.y = mu_ws[basei + (lowHalf ? 2 : 3)];
            }
            acc[j] = __builtin_amdgcn_wmma_f32_16x16x4_f32(
                false, a, false, bf, (short)0, acc[j], false, false);
        }
        __syncthreads();   // done reading `cur` before it is refilled
    }

    // ---- finalize: emis = om - 0.5*o2 - 0.5*m2, store [p][t][s] (s<8 only)
    #pragma unroll
    for (int j = 0; j < 8; ++j) {
        const int px = wave * 8 + j;
        const int p  = b * HW_ + px;
        // combine the two k-halves of the per-t |obs|^2 partials
        const float o2f = o2p[j] + __shfl_xor(o2p[j], 16);
        float m2v = 0.0f;
        if (n < S_) m2v = m2_ws[p * S_ + n];
        #pragma unroll
        for (int r = 0; r < 8; ++r) {
            const int tloc = r + (lowHalf ? 0 : 8);    // D layout: M = r (+8 hi half)
            const float o2r = __shfl(o2f, tloc);       // lane tloc holds o2[tloc]
            if (n < S_)
                emis_ws[(p * T_ + t0 + tloc) * S_ + n] =
                    acc[j][r] - 0.5f * (o2r + m2v);
        }
    }
}

// ---------------------------------------------------------------------------
// Kernel 3: alpha scan + fused outputs.
//   8 lanes per pixel (lane%8 = state), 4 pixels per wave, 8 waves per block,
//   8 blocks => 256 pixels.  Per step: 8x8 matvec via registers + shuffles.
// ---------------------------------------------------------------------------
__global__ void hsmm_scan(const float* __restrict__ trans_ws,   // [p][from][to]
                          const float* __restrict__ emis_ws,    // [p][t][s]
                          const float* __restrict__ dur_out,    // (B,S,D,H,W)
                          const float* __restrict__ initial_state, // (S,)
                          float* __restrict__ state_probs,      // (B,S,T,H,W)
                          float* __restrict__ loglik)           // (B,H,W)
{
    const int tid  = threadIdx.x;
    const int wave = tid >> 5;
    const int lane = tid & 31;
    const int s    = lane & 7;
    const int base = lane & 24;                 // first lane of my 8-lane group
    const int p    = blockIdx.x * 32 + wave * 4 + (lane >> 3);
    const int b    = p >> 6;
    const int hw   = p & 63;

    // softmax(initial_state) across the 8-lane group
    float v = initial_state[s];
    float m = v;
    m = fmaxf(m, __shfl_xor(m, 1));
    m = fmaxf(m, __shfl_xor(m, 2));
    m = fmaxf(m, __shfl_xor(m, 4));
    float e = __expf(v - m);
    float sum = e;
    sum += __shfl_xor(sum, 1);
    sum += __shfl_xor(sum, 2);
    sum += __shfl_xor(sum, 4);
    const float pi = e / sum;

    // transition column slice for my destination state s: trans[q][s], q=0..7
    float tc[S_];
    #pragma unroll
    for (int q = 0; q < S_; ++q)
        tc[q] = trans_ws[p * (S_ * S_) + q * S_ + s];

    const float dur0 = dur_out[((b * S_ + s) * D_ + 0) * HW_ + hw];
    const float* em = emis_ws + (size_t)p * T_ * S_;

    // t = 0
    float x = pi * em[s] * dur0;

    float a[S_];
    #pragma unroll
    for (int q = 0; q < S_; ++q) a[q] = __shfl(x, base + q);

    {
        float den = 1e-10f;
        #pragma unroll
        for (int q = 0; q < S_; ++q) den += a[q];
        state_probs[((size_t)(b * S_ + s) * T_ + 0) * HW_ + hw] = x / den;
    }

    for (int t = 1; t < T_; ++t) {
        // pull the emission stream ahead of the serial chain (L2-resident)
        if (((t & 15) == 0) && (t + 256 < T_))
            __builtin_prefetch(em + (t + 256) * S_, 0, 0);

        const float et = em[t * S_ + s];
        float acc = 0.0f;
        #pragma unroll
        for (int q = 0; q < S_; ++q) acc = fmaf(a[q], tc[q], acc);
        x = acc * et;
        #pragma unroll
        for (int q = 0; q < S_; ++q) a[q] = __shfl(x, base + q);
        float den = 1e-10f;
        #pragma unroll
        for (int q = 0; q < S_; ++q) den += a[q];
        state_probs[((size_t)(b * S_ + s) * T_ + t) * HW_ + hw] = x / den;
    }

    // log_likelihood = logsumexp over states of raw alpha at t = T-1
    if (s == 0) {
        float mm = a[0];
        #pragma unroll
        for (int q = 1; q < S_; ++q) mm = fmaxf(mm, a[q]);
        float se = 0.0f;
        #pragma unroll
        for (int q = 0; q < S_; ++q) se += __expf(a[q] - mm);
        loglik[b * HW_ + hw] = mm + __logf(se);
    }
}

// ---------------------------------------------------------------------------
extern "C" void kernel_launch(void* const* d_in, const int* in_sizes, int n_in,
                              void* d_out, int out_size, void* d_ws, size_t ws_size,
                              hipStream_t stream) {
    const float* features      = (const float*)d_in[0];
    const float* observations  = (const float*)d_in[1];
    const float* trans_w       = (const float*)d_in[2];
    const float* trans_b       = (const float*)d_in[3];
    const float* dur_w         = (const float*)d_in[4];
    const float* dur_b         = (const float*)d_in[5];
    const float* emis_w        = (const float*)d_in[6];
    const float* emis_b        = (const float*)d_in[7];
    const float* initial_state = (const float*)d_in[8];

    float* out   = (float*)d_out;
    float* sp    = out;                                   // (B,S,T,H,W)
    float* ll    = out + (size_t)B_ * S_ * T_ * HW_;      // (B,H,W)
    float* durp  = ll + (size_t)B_ * HW_;                 // (B,S,D,H,W)

    float* ws       = (float*)d_ws;
    float* trans_ws = ws;                                              // 16384
    float* mu_ws    = trans_ws + (size_t)P_ * S_ * S_;                 // 131072
    float* m2_ws    = mu_ws + (size_t)P_ * S_ * C_;                    // 2048
    float* emis_ws  = m2_ws + (size_t)P_ * S_;                         // 4.19M

    hipLaunchKernelGGL(hsmm_precompute, dim3(P_), dim3(64), 0, stream,
                       features, trans_w, trans_b, dur_w, dur_b, emis_w, emis_b,
                       durp, trans_ws, mu_ws, m2_ws);

    hipLaunchKernelGGL(hsmm_emission, dim3(T_ / 16, B_), dim3(256), 0, stream,
                       observations, mu_ws, m2_ws, emis_ws);

    hipLaunchKernelGGL(hsmm_scan, dim3(P_ / 32), dim3(256), 0, stream,
                       trans_ws, emis_ws, durp, initial_state, sp, ll);
}